// FISTA_22728966931068
// MI455X (gfx1250) — compile-verified
//
#include <hip/hip_runtime.h>
#include <hip/hip_bf16.h>

typedef float v2f __attribute__((ext_vector_type(2)));
typedef float v8f __attribute__((ext_vector_type(8)));

#define T_DIM   1024
#define NP_DIM  2048
#define B_DIM   16
#define J_DIM   64
#define TOTAL   (B_DIM * NP_DIM * J_DIM)   // 2,097,152
#define N_ITERS 25
#define LAMBDA_INIT 0.1f
#define TOL_SQ  (1e-6f * 1e-6f)
#define EPS_W   1e-10f

// ---------------------------------------------------------------------------
// Generic f32 WMMA GEMM: C[M x N] = A[M x K] * B[K x N], batched over
// blockIdx.y, N-tiled over blockIdx.z. Workgroup = 256 threads = 8 waves,
// computes a 128x64 output tile; each wave owns one 16-row strip x 64 cols
// (4 x V_WMMA_F32_16X16X4_F32 accumulators). K staged through LDS in tiles
// of 32 with register double-buffering of the global fetch.
// B is stored TRANSPOSED in LDS (sBt[n][k]) so both A and B fragments are
// consecutive-word pairs (single ds_load_2addr_b32 each). Fragment reads are
// software-pipelined one K-step ahead through distinct registers, and use
// per-nt base pointers so all DS offsets fit the immediate fields.
// aTrans=1 means A(m,k) = Amem[k*lda + m] (used for D^T).
// ---------------------------------------------------------------------------
__global__ __launch_bounds__(256)
void fista_wmma_gemm(const float* __restrict__ A, const float* __restrict__ Bm,
                     float* __restrict__ Cm,
                     int M, int K, int ldb, int ldc,
                     int aTrans, int lda,
                     long bBatchStride, long cBatchStride) {
  __shared__ float sA[128][33];   // [m][k], padded
  __shared__ float sBt[64][33];   // [n][k], padded (transposed B tile)

  const int tid   = threadIdx.x;
  const int lane  = tid & 31;
  const int wave  = tid >> 5;           // 0..7 -> M sub-strip
  const int lo    = lane & 15;
  const int hi    = lane >> 4;          // 0/1 -> K half selector per ISA layout
  const int mBase = blockIdx.x * 128;
  const int nBase = blockIdx.z * 64;

  const float* Bp = Bm + (long)blockIdx.y * bBatchStride;
  float*       Cp = Cm + (long)blockIdx.y * cBatchStride;

  v8f acc[4];
#pragma unroll
  for (int nt = 0; nt < 4; ++nt)
#pragma unroll
    for (int r = 0; r < 8; ++r) acc[nt][r] = 0.0f;

  float  aR[16];
  float4 bR[2];

  // ---- fetch tile at k into registers ----
  auto fetch = [&](int k) {
    if (aTrans) {
#pragma unroll
      for (int i = 0; i < 16; ++i) {
        const int lin = tid + i * 256;            // 0..4095
        aR[i] = A[(long)(k + (lin >> 7)) * lda + (mBase + (lin & 127))];
      }
    } else {
#pragma unroll
      for (int f = 0; f < 4; ++f) {
        const int fid = tid + f * 256;            // 0..1023
        const float4 v = *(const float4*)(
            A + (long)(mBase + (fid >> 3)) * K + k + ((fid & 7) << 2));
        aR[f * 4 + 0] = v.x; aR[f * 4 + 1] = v.y;
        aR[f * 4 + 2] = v.z; aR[f * 4 + 3] = v.w;
      }
    }
#pragma unroll
    for (int f = 0; f < 2; ++f) {
      const int fid = tid + f * 256;              // 0..511
      bR[f] = *(const float4*)(
          Bp + (long)(k + (fid >> 4)) * ldb + nBase + ((fid & 15) << 2));
    }
  };

  // ---- commit registers to LDS ----
  auto commit = [&]() {
    if (aTrans) {
#pragma unroll
      for (int i = 0; i < 16; ++i) {
        const int lin = tid + i * 256;
        sA[lin & 127][lin >> 7] = aR[i];
      }
    } else {
#pragma unroll
      for (int f = 0; f < 4; ++f) {
        const int fid = tid + f * 256;
        const int pl = fid >> 3, k4 = (fid & 7) << 2;
        sA[pl][k4 + 0] = aR[f * 4 + 0];
        sA[pl][k4 + 1] = aR[f * 4 + 1];
        sA[pl][k4 + 2] = aR[f * 4 + 2];
        sA[pl][k4 + 3] = aR[f * 4 + 3];
      }
    }
#pragma unroll
    for (int f = 0; f < 2; ++f) {
      const int fid = tid + f * 256;
      const int kl = fid >> 4, n4 = (fid & 15) << 2;
      sBt[n4 + 0][kl] = bR[f].x;
      sBt[n4 + 1][kl] = bR[f].y;
      sBt[n4 + 2][kl] = bR[f].z;
      sBt[n4 + 3][kl] = bR[f].w;
    }
  };

  // per-fragment base pointers: all inner-loop DS offsets are kk*4 <= 124 B,
  // encodable as ds_load_2addr immediate offsets (no per-load address VALU)
  const float* aP  = &sA [wave * 16 + lo][2 * hi];
  const float* bP0 = &sBt[ 0 + lo][2 * hi];
  const float* bP1 = &sBt[16 + lo][2 * hi];
  const float* bP2 = &sBt[32 + lo][2 * hi];
  const float* bP3 = &sBt[48 + lo][2 * hi];

  auto ldfrag = [&](int kk, v2f& a, v2f* b) {
    a.x    = aP [kk]; a.y    = aP [kk + 1];
    b[0].x = bP0[kk]; b[0].y = bP0[kk + 1];
    b[1].x = bP1[kk]; b[1].y = bP1[kk + 1];
    b[2].x = bP2[kk]; b[2].y = bP2[kk + 1];
    b[3].x = bP3[kk]; b[3].y = bP3[kk + 1];
  };

  fetch(0);
  commit();
  __syncthreads();

  for (int k0 = 0; k0 < K; k0 += 32) {
    const int kn = k0 + 32;
    if (kn < K) fetch(kn);   // overlap next-tile global loads with WMMAs

    v2f aC, aN;
    v2f bC[4], bN[4];
    ldfrag(0, aC, bC);
#pragma unroll
    for (int kk = 0; kk < 32; kk += 4) {
      if (kk + 4 < 32) ldfrag(kk + 4, aN, bN);   // pipeline next fragments
#pragma unroll
      for (int nt = 0; nt < 4; ++nt)
        acc[nt] = __builtin_amdgcn_wmma_f32_16x16x4_f32(
            false, aC, false, bC[nt], (short)0, acc[nt], false, false);
      aC = aN;                                   // renamed by full unroll
#pragma unroll
      for (int nt = 0; nt < 4; ++nt) bC[nt] = bN[nt];
    }

    if (kn < K) {
      __syncthreads();       // all LDS reads of this tile done
      commit();
      __syncthreads();       // new tile visible
    }
  }

  // store: C/D layout -> VGPR r: lanes 0-15 M=r, lanes 16-31 M=r+8
#pragma unroll
  for (int nt = 0; nt < 4; ++nt)
#pragma unroll
    for (int r = 0; r < 8; ++r) {
      const int row = mBase + wave * 16 + r + 8 * hi;
      const int col = nBase + nt * 16 + lo;
      Cp[(long)row * ldc + col] = acc[nt][r];
    }
}

// ---------------------------------------------------------------------------
// Power iteration for L = 1 / lambda_max(DtD). Single workgroup of 1024.
// Uses symmetry of DtD for coalesced column-major reads:
//   y[r] = sum_q DtD[q][r] * v[q]
// ---------------------------------------------------------------------------
__global__ __launch_bounds__(1024)
void fista_power_iter(const float* __restrict__ DtD, float* __restrict__ scal) {
  __shared__ float v[NP_DIM];
  __shared__ float red[1024];
  const int t = threadIdx.x;
  v[t] = 1.0f;
  v[t + 1024] = 1.0f;
  __syncthreads();
  float norm = 1.0f;
  for (int it = 0; it < 32; ++it) {
    float y0 = 0.0f, y1 = 0.0f;
    for (int q = 0; q < NP_DIM; ++q) {
      const float vq = v[q];
      const float* row = DtD + (long)q * NP_DIM;
      y0 = fmaf(row[t], vq, y0);
      y1 = fmaf(row[t + 1024], vq, y1);
    }
    red[t] = y0 * y0 + y1 * y1;
    __syncthreads();
    for (int s = 512; s > 0; s >>= 1) {
      if (t < s) red[t] += red[t + s];
      __syncthreads();
    }
    norm = sqrtf(red[0]);
    __syncthreads();                 // everyone read red[0] before reuse
    const float inv = 1.0f / norm;
    v[t] = y0 * inv;
    v[t + 1024] = y1 * inv;
    __syncthreads();
  }
  if (t == 0) scal[0] = 1.0f / norm;   // L = 1 / spectral norm
}

// ---------------------------------------------------------------------------
// Elementwise FISTA step: C_next = soft(Cp - L*(G - DtY), L*lam)
// G is overwritten in place with C_next; squared Frobenius diff accumulated
// into scal[3] via per-block reduction + one atomicAdd.
// scal layout: [0]=L [1]=t [2]=done [3]=sqdiff [4]=t_const [5]=doStep
// ---------------------------------------------------------------------------
__global__ __launch_bounds__(256)
void fista_update(const float* __restrict__ Cc, const float* __restrict__ Cp,
                  float* __restrict__ G, const float* __restrict__ DtY,
                  const float* __restrict__ lamArr, int useLamArr,
                  float* __restrict__ scal) {
  __shared__ float red[256];
  const float L = scal[0];
  const int base = blockIdx.x * 1024 + threadIdx.x;
  float local = 0.0f;
#pragma unroll
  for (int u = 0; u < 4; ++u) {
    const int i = base + u * 256;
    const float lam = useLamArr ? lamArr[i] : LAMBDA_INIT;
    const float grad = G[i] - DtY[i];
    const float x = Cp[i] - L * grad;
    const float ax = fabsf(x) - L * lam;
    const float cn = (ax > 0.0f) ? copysignf(ax, x) : 0.0f;
    G[i] = cn;
    const float d = cn - Cc[i];
    local += d * d;
  }
  red[threadIdx.x] = local;
  __syncthreads();
  for (int s = 128; s > 0; s >>= 1) {
    if (threadIdx.x < s) red[threadIdx.x] += red[threadIdx.x + s];
    __syncthreads();
  }
  if (threadIdx.x == 0) atomicAdd(&scal[3], red[0]);
}

__global__ void fista_control(float* __restrict__ scal) {
  const bool small  = scal[3] < TOL_SQ;
  const bool done   = scal[2] != 0.0f;
  const bool doStep = !(done || small);
  if (doStep) {
    const float t  = scal[1];
    const float tn = 0.5f * (1.0f + sqrtf(1.0f + 4.0f * t * t));
    scal[4] = (t - 1.0f) / tn;
    scal[1] = tn;
  }
  scal[5] = doStep ? 1.0f : 0.0f;
  scal[2] = (done || small) ? 1.0f : 0.0f;
  scal[3] = 0.0f;                     // reset accumulator for next iteration
}

__global__ __launch_bounds__(256)
void fista_apply(float* __restrict__ Cc, float* __restrict__ Cp,
                 const float* __restrict__ Cn, const float* __restrict__ scal) {
  if (scal[5] == 0.0f) return;        // state frozen (scan's where())
  const float tc = scal[4];
  const int base = blockIdx.x * 1024 + threadIdx.x;
#pragma unroll
  for (int u = 0; u < 4; ++u) {
    const int i = base + u * 256;
    const float cn = Cn[i];
    const float cc = Cc[i];
    Cc[i] = cn;
    Cp[i] = cn + tc * (cn - cc);
  }
}

// One block per (b,j) column: L2-normalize w = 1/(|Cc|+eps) over Np, then
// lam = LAMBDA_INIT * w_normalized * Np.
__global__ __launch_bounds__(256)
void fista_reweight(const float* __restrict__ Cc, float* __restrict__ lam) {
  __shared__ float red[256];
  const int col = blockIdx.x;                 // b*64 + j
  const int b = col >> 6, j = col & 63;
  const float* base = Cc + (long)b * NP_DIM * J_DIM + j;
  float local = 0.0f;
  for (int p = threadIdx.x; p < NP_DIM; p += 256) {
    const float w = 1.0f / (fabsf(base[(long)p * J_DIM]) + EPS_W);
    local += w * w;
  }
  red[threadIdx.x] = local;
  __syncthreads();
  for (int s = 128; s > 0; s >>= 1) {
    if (threadIdx.x < s) red[threadIdx.x] += red[threadIdx.x + s];
    __syncthreads();
  }
  const float inv = 1.0f / sqrtf(red[0]);
  for (int p = threadIdx.x; p < NP_DIM; p += 256) {
    const float w = 1.0f / (fabsf(base[(long)p * J_DIM]) + EPS_W);
    lam[(long)b * NP_DIM * J_DIM + (long)p * J_DIM + j] =
        LAMBDA_INIT * (w * inv) * (float)NP_DIM;
  }
}

__global__ void fista_init_scal(float* __restrict__ scal, int phase) {
  if (phase == 0) scal[1] = 1.0f;   // t carries over into phase 2
  scal[2] = 0.0f;                   // done
  scal[3] = 0.0f;                   // sqdiff
  scal[4] = 0.0f;
  scal[5] = 0.0f;
}

// ---------------------------------------------------------------------------
extern "C" void kernel_launch(void* const* d_in, const int* in_sizes, int n_in,
                              void* d_out, int out_size, void* d_ws, size_t ws_size,
                              hipStream_t stream) {
  const float* Y = (const float*)d_in[0];   // [16, 1024, 64]
  const float* D = (const float*)d_in[1];   // [1024, 2048]
  float* Cc = (float*)d_out;                // [16, 2048, 64]

  char* ws = (char*)d_ws;
  float* DtD = (float*)ws;  ws += (size_t)NP_DIM * NP_DIM * 4;   // 16 MB
  float* DtY = (float*)ws;  ws += (size_t)TOTAL * 4;             //  8 MB
  float* Cp  = (float*)ws;  ws += (size_t)TOTAL * 4;             //  8 MB
  float* G   = (float*)ws;  ws += (size_t)TOTAL * 4;             //  8 MB (also C_next)
  float* lam = (float*)ws;  ws += (size_t)TOTAL * 4;             //  8 MB
  float* scal = (float*)ws;

  hipMemsetAsync(Cc, 0, (size_t)TOTAL * 4, stream);
  hipMemsetAsync(Cp, 0, (size_t)TOTAL * 4, stream);
  fista_init_scal<<<1, 1, 0, stream>>>(scal, 0);

  // DtD = D^T D  : M=N=2048, K=1024, A transposed view of D
  dim3 gDtD(NP_DIM / 128, 1, NP_DIM / 64);
  fista_wmma_gemm<<<gDtD, 256, 0, stream>>>(D, D, DtD,
      NP_DIM, T_DIM, NP_DIM, NP_DIM, /*aTrans=*/1, NP_DIM, 0L, 0L);

  // DtY[b] = D^T Y[b] : M=2048, N=64, K=1024, batched over 16
  dim3 gDtY(NP_DIM / 128, B_DIM, 1);
  fista_wmma_gemm<<<gDtY, 256, 0, stream>>>(D, Y, DtY,
      NP_DIM, T_DIM, J_DIM, J_DIM, /*aTrans=*/1, NP_DIM,
      (long)T_DIM * J_DIM, (long)NP_DIM * J_DIM);

  // L = 1 / ||DtD||_2 via power iteration
  fista_power_iter<<<1, 1024, 0, stream>>>(DtD, scal);

  dim3 gIter(NP_DIM / 128, B_DIM, 1);
  for (int phase = 0; phase < 2; ++phase) {
    if (phase == 1) {
      fista_reweight<<<B_DIM * J_DIM, 256, 0, stream>>>(Cc, lam);
      fista_init_scal<<<1, 1, 0, stream>>>(scal, 1);
    }
    for (int it = 0; it < N_ITERS; ++it) {
      // G[b] = DtD @ Cp[b]
      fista_wmma_gemm<<<gIter, 256, 0, stream>>>(DtD, Cp, G,
          NP_DIM, NP_DIM, J_DIM, J_DIM, /*aTrans=*/0, NP_DIM,
          (long)NP_DIM * J_DIM, (long)NP_DIM * J_DIM);
      fista_update<<<TOTAL / 1024, 256, 0, stream>>>(Cc, Cp, G, DtY,
          lam, phase, scal);
      fista_control<<<1, 1, 0, stream>>>(scal);
      fista_apply<<<TOTAL / 1024, 256, 0, stream>>>(Cc, Cp, G, scal);
    }
  }
}